// SelfAttention_31791347925883
// MI455X (gfx1250) — compile-verified
//
#include <hip/hip_runtime.h>

// CDNA5 / gfx1250 self-attention forward, fp32 end-to-end.
// All three matmul stages use V_WMMA_F32_16X16X4_F32 (full-precision fp32
// matrix pipe) so numerics match the fp32 reference.

typedef float v2f __attribute__((ext_vector_type(2)));
typedef float v8f __attribute__((ext_vector_type(8)));

#define T_SEQ 2048
#define HEADS 4
#define DH    64
#define C_DIM 256

__device__ __forceinline__ v8f wmma4(v2f a, v2f b, v8f c) {
  // D = A(16x4,f32) * B(4x16,f32) + C(16x16,f32)
  return __builtin_amdgcn_wmma_f32_16x16x4_f32(false, a, false, b, (short)0, c,
                                               false, false);
}

// ---------------------------------------------------------------------------
// Kernel 1: qkv = x @ W_qkv + b_qkv ; scatter to Q/K/V laid out [B*H][T][DH]
// One wave computes one 16x16 output tile.
// ---------------------------------------------------------------------------
__global__ void qkv_proj_kernel(const float* __restrict__ X,
                                const float* __restrict__ W,
                                const float* __restrict__ bias,
                                float* __restrict__ Qw,
                                float* __restrict__ Kw,
                                float* __restrict__ Vw,
                                int mtiles) {
  const int lane = threadIdx.x & 31;
  const int lo = lane & 15;
  const int hi = lane >> 4;                       // 0 or 1 (lane half)
  const int ntiles = (3 * C_DIM) / 16;            // 48
  int tile = (int)((blockIdx.x * blockDim.x + threadIdx.x) >> 5);
  if (tile >= mtiles * ntiles) return;            // wave-uniform
  const int mt = tile / ntiles;
  const int nt = tile - mt * ntiles;
  const int m0 = mt * 16, n0 = nt * 16;

  const float* arow = X + (size_t)(m0 + lo) * C_DIM;  // A: lane = row M
  const float* wcol = W + (n0 + lo);                  // B: lane = col N
  v8f c = {};
#pragma unroll 8
  for (int kk = 0; kk < C_DIM; kk += 4) {
    const int ka = kk + hi * 2;                   // K split across lane halves
    v2f a, b;
    a.x = arow[ka];
    a.y = arow[ka + 1];
    b.x = wcol[(size_t)ka * (3 * C_DIM)];
    b.y = wcol[(size_t)(ka + 1) * (3 * C_DIM)];
    c = wmma4(a, b, c);
  }

  // Scatter to Q/K/V [B*H][T][DH]. which/h are wave-uniform (16 | 64 | 256).
  const int cg = n0 + lo;                         // global qkv column (0..767)
  const int which = cg >> 8;                      // 0=Q 1=K 2=V
  const int within = cg & 255;
  const int h = within >> 6;
  const int dd = within & 63;
  const float bv = bias[cg];
  float* dst = (which == 0) ? Qw : ((which == 1) ? Kw : Vw);
#pragma unroll
  for (int r = 0; r < 8; ++r) {
    const int row = m0 + r + hi * 8;              // flat B*T row
    const int bb = row >> 11;                     // / T_SEQ
    const int tt = row & (T_SEQ - 1);
    dst[(((size_t)(bb * HEADS + h)) * T_SEQ + tt) * DH + dd] = c[r] + bv;
  }
}

// ---------------------------------------------------------------------------
// Kernel 2: causal flash attention, one wave per (b,h, 16-query tile).
//   S^T = K * Q^T   (C-layout: lane = query, vgpr/half = key)
//   online softmax per query column (per-lane + one shfl_xor(16))
//   O^T += V^T * P^T (P^T rebuilt as B-operand via half-swap shuffles)
// ---------------------------------------------------------------------------
__global__ void attn_kernel(const float* __restrict__ Q,
                            const float* __restrict__ K,
                            const float* __restrict__ V,
                            float* __restrict__ O,
                            int nwaves) {
  const int lane = threadIdx.x & 31;
  const int lo = lane & 15;
  const int hi = lane >> 4;
  int wave = (int)((blockIdx.x * blockDim.x + threadIdx.x) >> 5);
  if (wave >= nwaves) return;                     // wave-uniform

  const int qtiles = T_SEQ / 16;                  // 128
  const int qt = wave % qtiles;
  const int bh = wave / qtiles;                   // b*HEADS + h
  const int q0 = qt * 16;
  const float* Qh = Q + (size_t)bh * T_SEQ * DH;
  const float* Kh = K + (size_t)bh * T_SEQ * DH;
  const float* Vh = V + (size_t)bh * T_SEQ * DH;

  // Preload Q tile as the B-operand of S^T = K * Q^T, scale folded in.
  // scale = 1/sqrt(64) * log2(e)  (softmax done in exp2 domain -> v_exp_f32)
  const float qs = 0.125f * 1.44269504088896340736f;
  v2f qreg[16];
  {
    const float* qrow = Qh + (size_t)(q0 + lo) * DH;
#pragma unroll
    for (int i = 0; i < 16; ++i) {
      qreg[i].x = qrow[i * 4 + hi * 2] * qs;
      qreg[i].y = qrow[i * 4 + hi * 2 + 1] * qs;
    }
  }

  float m_run = -__builtin_inff();
  float l_run = 0.0f;
  v8f o0 = {}, o1 = {}, o2 = {}, o3 = {};         // O^T accumulators (4 d-chunks)

  for (int kt0 = 0; kt0 <= q0; kt0 += 16) {       // causal: skip kt0 > q0
    // ---- S^T tile: A = K rows, B = Q^T (preloaded) ----
    v8f s = {};
    const float* krow = Kh + (size_t)(kt0 + lo) * DH;
#pragma unroll
    for (int i = 0; i < 16; ++i) {
      v2f a;
      a.x = krow[i * 4 + hi * 2];
      a.y = krow[i * 4 + hi * 2 + 1];
      s = wmma4(a, qreg[i], s);
    }
    // diagonal tile: mask key > query (local: r + hi*8 > lo), per-lane select
    if (kt0 == q0) {
#pragma unroll
      for (int r = 0; r < 8; ++r)
        s[r] = (r + hi * 8 > lo) ? -__builtin_inff() : s[r];
    }

    // ---- online softmax (per query column) ----
    float tmax = s[0];
#pragma unroll
    for (int r = 1; r < 8; ++r) tmax = fmaxf(tmax, s[r]);
    tmax = fmaxf(tmax, __shfl_xor(tmax, 16, 32)); // combine lane halves
    const float m_new = fmaxf(m_run, tmax);
    const float alpha = __builtin_amdgcn_exp2f(m_run - m_new);
    float p[8];
    float ts = 0.0f;
#pragma unroll
    for (int r = 0; r < 8; ++r) {
      p[r] = __builtin_amdgcn_exp2f(s[r] - m_new);
      ts += p[r];
    }
    ts += __shfl_xor(ts, 16, 32);
    l_run = l_run * alpha + ts;
    m_run = m_new;
#pragma unroll
    for (int r = 0; r < 8; ++r) {
      o0[r] *= alpha; o1[r] *= alpha; o2[r] *= alpha; o3[r] *= alpha;
    }

    // ---- O^T += V^T * P^T ----
    // P^T is in C-layout: lane(q,hi) holds keys {hi*8..hi*8+7} in p[0..7].
    // B-operand for step kk needs keys {kk,kk+1} (hi=0) / {kk+2,kk+3} (hi=1):
    // fetch the cross-half pair with shfl_xor(16), select per half.
    const float* vbase = Vh + (size_t)kt0 * DH;
#pragma unroll
    for (int kk = 0; kk < 16; kk += 4) {
      v2f bop;
      if (kk < 8) {                               // kk = 0, 4
        const float sx = __shfl_xor(p[kk + 2], 16, 32);
        const float sy = __shfl_xor(p[kk + 3], 16, 32);
        bop.x = hi ? sx : p[kk];
        bop.y = hi ? sy : p[kk + 1];
      } else {                                    // kk = 8, 12
        const int j = kk - 8;
        const float sx = __shfl_xor(p[j], 16, 32);
        const float sy = __shfl_xor(p[j + 1], 16, 32);
        bop.x = hi ? p[j + 2] : sx;
        bop.y = hi ? p[j + 3] : sy;
      }
      const float* vk = vbase + (size_t)(kk + hi * 2) * DH + lo;
      v2f a;
      a.x = vk[0];       a.y = vk[DH];        o0 = wmma4(a, bop, o0);
      a.x = vk[16];      a.y = vk[DH + 16];   o1 = wmma4(a, bop, o1);
      a.x = vk[32];      a.y = vk[DH + 32];   o2 = wmma4(a, bop, o2);
      a.x = vk[48];      a.y = vk[DH + 48];   o3 = wmma4(a, bop, o3);
    }
  }

  // normalize + store to [B][T][H*DH] (head-major within the row)
  const float inv_l = 1.0f / l_run;
  const int b = bh / HEADS;
  const int h = bh - b * HEADS;
  float* orow = O + ((size_t)b * T_SEQ + q0 + lo) * C_DIM + h * DH + hi * 8;
#pragma unroll
  for (int r = 0; r < 8; ++r) {
    orow[r]      = o0[r] * inv_l;
    orow[16 + r] = o1[r] * inv_l;
    orow[32 + r] = o2[r] * inv_l;
    orow[48 + r] = o3[r] * inv_l;
  }
}

// ---------------------------------------------------------------------------
// Kernel 3: out = attn @ W_fc + b_fc   (one wave per 16x16 tile)
// ---------------------------------------------------------------------------
__global__ void out_proj_kernel(const float* __restrict__ A,
                                const float* __restrict__ W,
                                const float* __restrict__ bias,
                                float* __restrict__ Out,
                                int mtiles) {
  const int lane = threadIdx.x & 31;
  const int lo = lane & 15;
  const int hi = lane >> 4;
  const int ntiles = C_DIM / 16;                  // 16
  int tile = (int)((blockIdx.x * blockDim.x + threadIdx.x) >> 5);
  if (tile >= mtiles * ntiles) return;            // wave-uniform
  const int mt = tile / ntiles;
  const int nt = tile - mt * ntiles;
  const int m0 = mt * 16, n0 = nt * 16;

  const float* arow = A + (size_t)(m0 + lo) * C_DIM;
  const float* wcol = W + (n0 + lo);
  v8f c = {};
#pragma unroll 8
  for (int kk = 0; kk < C_DIM; kk += 4) {
    const int ka = kk + hi * 2;
    v2f a, b;
    a.x = arow[ka];
    a.y = arow[ka + 1];
    b.x = wcol[(size_t)ka * C_DIM];
    b.y = wcol[(size_t)(ka + 1) * C_DIM];
    c = wmma4(a, b, c);
  }
  const float bv = bias[n0 + lo];
#pragma unroll
  for (int r = 0; r < 8; ++r)
    Out[(size_t)(m0 + r + hi * 8) * C_DIM + n0 + lo] = c[r] + bv;
}

// ---------------------------------------------------------------------------
extern "C" void kernel_launch(void* const* d_in, const int* in_sizes, int n_in,
                              void* d_out, int out_size, void* d_ws, size_t ws_size,
                              hipStream_t stream) {
  const float* x     = (const float*)d_in[0];   // [B, T, 256]
  const float* W_qkv = (const float*)d_in[1];   // [256, 768]
  const float* b_qkv = (const float*)d_in[2];   // [768]
  const float* W_fc  = (const float*)d_in[3];   // [256, 256]
  const float* b_fc  = (const float*)d_in[4];   // [256]
  float* out = (float*)d_out;

  const int BT = in_sizes[0] / C_DIM;           // B*T
  const int B  = BT / T_SEQ;

  // workspace: Q, K, V each [B*H][T][DH], attn output [B][T][C]
  const size_t headElems = (size_t)B * HEADS * T_SEQ * DH;  // == BT*C_DIM
  float* Qw = (float*)d_ws;
  float* Kw = Qw + headElems;
  float* Vw = Kw + headElems;
  float* Aw = Vw + headElems;

  const int mtiles = BT / 16;

  {
    const int tiles = mtiles * ((3 * C_DIM) / 16);          // 16x16 tiles
    const int blocks = (tiles + 7) / 8;                     // 8 waves / block
    qkv_proj_kernel<<<blocks, 256, 0, stream>>>(x, W_qkv, b_qkv, Qw, Kw, Vw,
                                                mtiles);
  }
  {
    const int nwaves = B * HEADS * (T_SEQ / 16);
    const int blocks = (nwaves + 7) / 8;
    attn_kernel<<<blocks, 256, 0, stream>>>(Qw, Kw, Vw, Aw, nwaves);
  }
  {
    const int tiles = mtiles * (C_DIM / 16);
    const int blocks = (tiles + 7) / 8;
    out_proj_kernel<<<blocks, 256, 0, stream>>>(Aw, W_fc, b_fc, out, mtiles);
  }
}